// MnistNeuralDecoder_33285996544726
// MI455X (gfx1250) — compile-verified
//
#include <hip/hip_runtime.h>
#include <math.h>

// ---- problem constants (from reference) ----
#define GRID_N 64
#define HID    3
#define KS     11
#define PAD    5
#define BATCH  64
#define TSTEPS 32
#define NPIX   (GRID_N*GRID_N)          // 4096
#define KDIM   ((1+HID)*KS*KS)          // 484 (true K)
#define MPAD   16                       // 12 gate-channels padded to 16
#define FK     (GRID_N-27)              // 37
#define OUTW   (GRID_N-FK+1)            // 28
#define EPSLN  1e-5f

// Padded im2col K layout: each (ch,ky) row padded 11 -> 12 (zero weight in
// column 11) so the K loop tiles exactly into steps of 4 with constant offsets.
#define KROW   12                       // padded kx stride
#define KCH    (KS*KROW)                // 132 per channel
#define KPADT  (4*KCH)                  // 528 total padded K

typedef float v2f __attribute__((ext_vector_type(2)));
typedef float v8f __attribute__((ext_vector_type(8)));

// patch: 4 channels x 11 rows x 27 cols (16 px tile + 2*PAD halo + 1 pad col)
#define PATCH_W 27
#define PATCH_ELEMS (4*KS*PATCH_W)      // 1188
#define WAVES_PER_BLOCK 4

__device__ __forceinline__ float sigmoidf_(float x) {
    return 1.0f / (1.0f + __expf(-x));
}

// ---------------------------------------------------------------------------
// Gate convolution via im2col + V_WMMA_F32_16X16X4_F32.
// One wave32 per 16-pixel tile. All LDS offsets are compile-time immediates.
// ---------------------------------------------------------------------------
__global__ void __launch_bounds__(32*WAVES_PER_BLOCK)
convlstm_gates_wmma(const float* __restrict__ x,      // (B,T,1,64,64)
                    const float* __restrict__ Wg,     // (12,4,11,11) == (12,484)
                    const float* __restrict__ bg,     // (12)
                    const float* __restrict__ h,      // (B,3,64,64)
                    float* __restrict__ z,            // (B,12,64,64)
                    int t)
{
    __shared__ float Alds[MPAD * KPADT];                   // 16x528 weight panel
    __shared__ float patch[WAVES_PER_BLOCK][PATCH_ELEMS];  // per-wave im2col halo

    const int tid  = threadIdx.x;
    const int lane = tid & 31;
    const int wv   = tid >> 5;

    const int b    = blockIdx.x >> 6;        // 64 blocks per batch item
    const int tblk = blockIdx.x & 63;
    const int tile = tblk * WAVES_PER_BLOCK + wv;   // 0..255
    const int py   = tile >> 2;              // output row
    const int px0  = (tile & 3) << 4;        // output col base (16-wide)

    // ---- stage padded weight panel (zero in pad rows and pad kx column) ----
    for (int i = tid; i < MPAD * KPADT; i += 32*WAVES_PER_BLOCK) {
        int m  = i / KPADT;
        int r  = i - m * KPADT;
        int ch = r / KCH;
        int rr = r - ch * KCH;
        int ky = rr / KROW;
        int kx = rr - ky * KROW;
        float v = 0.0f;
        if (m < 4*HID && kx < KS)
            v = Wg[m * KDIM + ch * (KS*KS) + ky * KS + kx];
        Alds[i] = v;
    }

    // ---- stage this wave's input halo patch (zero-padded borders) ----
    const float* xt = x + ((size_t)b * TSTEPS + t) * NPIX;   // channel 0
    const float* hb = h + (size_t)b * HID * NPIX;            // channels 1..3
    for (int i = lane; i < PATCH_ELEMS; i += 32) {
        int ch = i / (KS * PATCH_W);
        int r  = (i % (KS * PATCH_W)) / PATCH_W;
        int cx = i % PATCH_W;
        int gy = py - PAD + r;
        int gx = px0 - PAD + cx;
        float v = 0.0f;
        if ((unsigned)gy < GRID_N && (unsigned)gx < GRID_N) {
            int off = gy * GRID_N + gx;
            v = (ch == 0) ? xt[off] : hb[(ch - 1) * NPIX + off];
        }
        patch[wv][i] = v;
    }

    // Pull next timestep's x rows toward L2 (global_prefetch_b8); the 192MB
    // L2 easily holds the full 33MB x tensor once warmed.
    if (t + 1 < TSTEPS) {
        const float* xn = x + ((size_t)b * TSTEPS + (t + 1)) * NPIX
                            + py * GRID_N + (lane << 1);
        __builtin_prefetch(xn, 0, 1);
    }

    __syncthreads();

    // ---- 16x16 tile GEMM: fully unrolled 4*11*3 = 132 WMMA K-steps ----
    const int m  = lane & 15;                 // A row held by this lane
    const int n  = lane & 15;                 // B/C column held by this lane
    const int kb = (lane >> 4) << 1;          // lanes 16..31 hold K=2,3
    const float* Arow = &Alds[m * KPADT] + kb;
    const float* P    = &patch[wv][n] + kb;

    v8f acc = {};
    #pragma unroll
    for (int ch = 0; ch < 4; ++ch) {
        #pragma unroll
        for (int ky = 0; ky < KS; ++ky) {
            const int abase = ch * KCH + ky * KROW;          // constant
            const int bbase = (ch * KS + ky) * PATCH_W;      // constant
            #pragma unroll
            for (int kxb = 0; kxb < KROW; kxb += 4) {
                v2f a, bf;
                a.x  = Arow[abase + kxb];
                a.y  = Arow[abase + kxb + 1];
                bf.x = P[bbase + kxb];
                bf.y = P[bbase + kxb + 1];
                acc = __builtin_amdgcn_wmma_f32_16x16x4_f32(
                        false, a, false, bf, (short)0, acc, false, false);
            }
        }
    }

    // ---- writeback: VGPR v, lane l -> (M = v + 8*(l>=16), N = l&15) ----
    const int mhi = (lane >> 4) << 3;
    float* zb = z + (size_t)b * (4*HID) * NPIX + py * GRID_N + px0 + n;
    #pragma unroll
    for (int v = 0; v < 8; ++v) {
        int mm = v + mhi;
        if (mm < 4*HID) {
            zb[(size_t)mm * NPIX] = acc[v] + bg[mm];
        }
    }
}

// ---------------------------------------------------------------------------
// Per-(batch, gate) LayerNorm statistics over HID*64*64 = 12288 elements.
// stats[bg*2] = mu, stats[bg*2+1] = rsqrt(var+eps)
// ---------------------------------------------------------------------------
__global__ void __launch_bounds__(256)
ln_stats(const float* __restrict__ z, float* __restrict__ stats)
{
    __shared__ float s_sum[256];
    __shared__ float s_sq[256];
    const int b = blockIdx.x >> 2;
    const int g = blockIdx.x & 3;
    const float* p = z + ((size_t)b * (4*HID) + g * HID) * NPIX;  // contiguous 12288
    const int N = HID * NPIX;

    float s = 0.f, s2 = 0.f;
    for (int i = threadIdx.x; i < N; i += 256) {
        float v = p[i];
        s += v; s2 += v * v;
    }
    s_sum[threadIdx.x] = s;
    s_sq[threadIdx.x]  = s2;
    __syncthreads();
    for (int off = 128; off > 0; off >>= 1) {
        if (threadIdx.x < off) {
            s_sum[threadIdx.x] += s_sum[threadIdx.x + off];
            s_sq[threadIdx.x]  += s_sq[threadIdx.x + off];
        }
        __syncthreads();
    }
    if (threadIdx.x == 0) {
        float mu  = s_sum[0] / (float)N;
        float var = s_sq[0] / (float)N - mu * mu;
        stats[blockIdx.x * 2 + 0] = mu;
        stats[blockIdx.x * 2 + 1] = rsqrtf(var + EPSLN);
    }
}

// ---------------------------------------------------------------------------
// LN apply + peephole gates + state update (one thread per (b,hid,pixel)).
// ---------------------------------------------------------------------------
__global__ void __launch_bounds__(256)
gate_update(const float* __restrict__ z,      // (B,12,4096)
            const float* __restrict__ stats,  // (B,4,2)
            const float* __restrict__ ln_w,   // (4,3,4096)
            const float* __restrict__ ln_b,   // (4,3,4096)
            const float* __restrict__ w_ci,   // (3,4096)
            const float* __restrict__ w_cf,
            const float* __restrict__ w_co,
            float* __restrict__ h,            // (B,3,4096)
            float* __restrict__ c)
{
    const int idx = blockIdx.x * 256 + threadIdx.x;   // B*3*4096 threads exactly
    const int b   = idx / (HID * NPIX);
    const int r   = idx - b * (HID * NPIX);
    const int hc  = r / NPIX;
    const int p   = r - hc * NPIX;

    const size_t zb = (size_t)b * (4*HID) * NPIX;
    float zn[4];
    #pragma unroll
    for (int g = 0; g < 4; ++g) {
        float mu   = stats[(b * 4 + g) * 2 + 0];
        float rstd = stats[(b * 4 + g) * 2 + 1];
        int   lidx = (g * HID + hc) * NPIX + p;
        float v    = z[zb + (size_t)(g * HID + hc) * NPIX + p];
        zn[g] = (v - mu) * rstd * ln_w[lidx] + ln_b[lidx];
    }

    const int pp = hc * NPIX + p;
    float cv = c[idx];
    float ig = sigmoidf_(zn[0] + w_ci[pp] * cv);
    float fg = sigmoidf_(zn[1] + w_cf[pp] * cv);
    float cn = fg * cv + ig * tanhf(zn[2]);
    float og = sigmoidf_(zn[3] + w_co[pp] * cn);
    c[idx] = cn;
    h[idx] = og * tanhf(cn);
}

__global__ void __launch_bounds__(256)
zero_init(float* __restrict__ p, int n)
{
    int i = blockIdx.x * 256 + threadIdx.x;
    if (i < n) p[i] = 0.0f;
}

// ---------------------------------------------------------------------------
// Final 37x37 valid conv: out (B,1,28,28). 0.2% of total FLOPs -> scalar.
// ---------------------------------------------------------------------------
__global__ void __launch_bounds__(256)
final_conv(const float* __restrict__ h,       // (B,3,64,64)
           const float* __restrict__ cw,      // (1,3,37,37)
           const float* __restrict__ cb,      // (1)
           float* __restrict__ out)           // (B,1,28,28)
{
    const int idx = blockIdx.x * 256 + threadIdx.x;   // B*28*28 threads exactly
    const int b  = idx / (OUTW * OUTW);
    const int r  = idx - b * (OUTW * OUTW);
    const int oy = r / OUTW;
    const int ox = r - oy * OUTW;

    float acc = cb[0];
    for (int ch = 0; ch < HID; ++ch) {
        const float* hp = h + ((size_t)b * HID + ch) * NPIX;
        const float* wp = cw + ch * FK * FK;
        for (int ky = 0; ky < FK; ++ky) {
            const float* hrow = hp + (oy + ky) * GRID_N + ox;
            const float* wrow = wp + ky * FK;
            #pragma unroll 4
            for (int kx = 0; kx < FK; ++kx)
                acc += hrow[kx] * wrow[kx];
        }
    }
    out[idx] = acc;
}

// ---------------------------------------------------------------------------
extern "C" void kernel_launch(void* const* d_in, const int* in_sizes, int n_in,
                              void* d_out, int out_size, void* d_ws, size_t ws_size,
                              hipStream_t stream) {
    (void)in_sizes; (void)n_in; (void)out_size; (void)ws_size;
    const float* x      = (const float*)d_in[0];
    const float* Wg     = (const float*)d_in[1];
    const float* bg     = (const float*)d_in[2];
    const float* w_ci   = (const float*)d_in[3];
    const float* w_cf   = (const float*)d_in[4];
    const float* w_co   = (const float*)d_in[5];
    const float* ln_w   = (const float*)d_in[6];
    const float* ln_b   = (const float*)d_in[7];
    const float* conv_w = (const float*)d_in[8];
    const float* conv_b = (const float*)d_in[9];
    float* out = (float*)d_out;

    // workspace layout (floats)
    float* ws    = (float*)d_ws;
    float* z     = ws;                                   // B*12*4096
    float* hbuf  = z + (size_t)BATCH * 4*HID * NPIX;     // B*3*4096
    float* cbuf  = hbuf + (size_t)BATCH * HID * NPIX;    // B*3*4096
    float* stats = cbuf + (size_t)BATCH * HID * NPIX;    // B*4*2

    const int state_n = 2 * BATCH * HID * NPIX;          // h and c contiguous
    zero_init<<<(state_n + 255) / 256, 256, 0, stream>>>(hbuf, state_n);

    for (int t = 0; t < TSTEPS; ++t) {
        convlstm_gates_wmma<<<BATCH * 64, 32 * WAVES_PER_BLOCK, 0, stream>>>(
            x, Wg, bg, hbuf, z, t);
        ln_stats<<<BATCH * 4, 256, 0, stream>>>(z, stats);
        gate_update<<<(BATCH * HID * NPIX) / 256, 256, 0, stream>>>(
            z, stats, ln_w, ln_b, w_ci, w_cf, w_co, hbuf, cbuf);
    }

    final_conv<<<(BATCH * OUTW * OUTW) / 256, 256, 0, stream>>>(
        hbuf, conv_w, conv_b, out);
}